// MyLoss_36309653520528
// MI455X (gfx1250) — compile-verified
//
#include <hip/hip_runtime.h>

// ---------------------------------------------------------------------------
// Top-k(|input-label|) masked MSE loss for MI455X (gfx1250, wave32).
// Radix-select on f32 bit patterns (3 data passes; passes 2/3 are L2-resident
// since 134MB < 192MB L2), with the masked sum-of-squares reductions done via
// V_WMMA_F32_16X16X4_F32 (A = ones => layout-independent reduction).
// ---------------------------------------------------------------------------

typedef float v2f __attribute__((ext_vector_type(2)));
typedef float v8f __attribute__((ext_vector_type(8)));

#define NSLICE   64        // B*C
#define NELEM    262144    // D*H*W
#define KSEL     131072u   // ceil(0.5*N)
#define WG_PER_SLICE 8
#define NTHREADS 256
#define CHUNK    (NELEM / WG_PER_SLICE)       // 32768 elems per block
#define ITERS    (CHUNK / (NTHREADS * 4))     // 32 float4 iters per thread

__device__ __forceinline__ float wmma_pair_reduce_finish(v8f acc) {
    return acc[0] + acc[1] + acc[2] + acc[3] + acc[4] + acc[5] + acc[6] + acc[7];
}

// ---------------- Pass 1: coarse histogram of bits[30:19] ------------------
__global__ void __launch_bounds__(NTHREADS)
k_hist1(const float* __restrict__ in, const float* __restrict__ lab,
        unsigned* __restrict__ hist1) {
    __shared__ unsigned lh[4096];
    const int slice = blockIdx.x >> 3;
    const int chunk = blockIdx.x & 7;
    for (int i = threadIdx.x; i < 4096; i += NTHREADS) lh[i] = 0u;
    __syncthreads();

    const size_t base = (size_t)slice * NELEM + (size_t)chunk * CHUNK;
    const float4* in4 = (const float4*)(in + base);
    const float4* lb4 = (const float4*)(lab + base);

    for (int it = 0; it < ITERS; ++it) {
        const int idx = it * NTHREADS + threadIdx.x;
        float4 a = in4[idx];
        float4 b = lb4[idx];
        unsigned u0 = __float_as_uint(fabsf(a.x - b.x));
        unsigned u1 = __float_as_uint(fabsf(a.y - b.y));
        unsigned u2 = __float_as_uint(fabsf(a.z - b.z));
        unsigned u3 = __float_as_uint(fabsf(a.w - b.w));
        atomicAdd(&lh[u0 >> 19], 1u);
        atomicAdd(&lh[u1 >> 19], 1u);
        atomicAdd(&lh[u2 >> 19], 1u);
        atomicAdd(&lh[u3 >> 19], 1u);
    }
    __syncthreads();
    unsigned* g = hist1 + (size_t)slice * 4096;
    for (int i = threadIdx.x; i < 4096; i += NTHREADS) {
        unsigned c = lh[i];
        if (c) atomicAdd(&g[i], c);
    }
}

// ---------------- Scan 1: find cut bucket b1 and residual need -------------
__global__ void k_scan1(const unsigned* __restrict__ hist1,
                        unsigned* __restrict__ b1, unsigned* __restrict__ need1) {
    const int s = threadIdx.x;  // launched with 64 threads
    const unsigned* h = hist1 + (size_t)s * 4096;
    unsigned cum = 0;
    for (int b = 4095; b >= 0; --b) {
        unsigned c = h[b];
        if (cum + c >= KSEL) { b1[s] = (unsigned)b; need1[s] = KSEL - cum; return; }
        cum += c;
    }
}

// ------- Pass 2: WMMA sum of v^2 above b1; mid histogram at b1 -------------
__global__ void __launch_bounds__(NTHREADS)
k_pass2(const float* __restrict__ in, const float* __restrict__ lab,
        const unsigned* __restrict__ b1arr,
        unsigned* __restrict__ hist2, double* __restrict__ sumsq) {
    __shared__ unsigned lh[4096];
    __shared__ float part[NTHREADS];
    const int slice = blockIdx.x >> 3;
    const int chunk = blockIdx.x & 7;
    const unsigned b1 = b1arr[slice];
    for (int i = threadIdx.x; i < 4096; i += NTHREADS) lh[i] = 0u;
    __syncthreads();

    const size_t base = (size_t)slice * NELEM + (size_t)chunk * CHUNK;
    const float4* in4 = (const float4*)(in + base);
    const float4* lb4 = (const float4*)(lab + base);

    v8f acc = {0.f, 0.f, 0.f, 0.f, 0.f, 0.f, 0.f, 0.f};
    const v2f ones = {1.0f, 1.0f};

    for (int it = 0; it < ITERS; ++it) {
        const int idx = it * NTHREADS + threadIdx.x;
        float4 a = in4[idx];
        float4 b = lb4[idx];
        float d0 = fabsf(a.x - b.x), d1 = fabsf(a.y - b.y);
        float d2 = fabsf(a.z - b.z), d3 = fabsf(a.w - b.w);
        unsigned u0 = __float_as_uint(d0), u1 = __float_as_uint(d1);
        unsigned u2 = __float_as_uint(d2), u3 = __float_as_uint(d3);
        unsigned k0 = u0 >> 19, k1 = u1 >> 19, k2 = u2 >> 19, k3 = u3 >> 19;
        // Masked squares (selects, no divergence -> EXEC all ones at WMMA).
        v2f p = { (k0 > b1) ? d0 * d0 : 0.0f, (k1 > b1) ? d1 * d1 : 0.0f };
        v2f q = { (k2 > b1) ? d2 * d2 : 0.0f, (k3 > b1) ? d3 * d3 : 0.0f };
        acc = __builtin_amdgcn_wmma_f32_16x16x4_f32(false, ones, false, p,
                                                    (short)0, acc, false, false);
        acc = __builtin_amdgcn_wmma_f32_16x16x4_f32(false, ones, false, q,
                                                    (short)0, acc, false, false);
        if (k0 == b1) atomicAdd(&lh[(u0 >> 7) & 0xFFFu], 1u);
        if (k1 == b1) atomicAdd(&lh[(u1 >> 7) & 0xFFFu], 1u);
        if (k2 == b1) atomicAdd(&lh[(u2 >> 7) & 0xFFFu], 1u);
        if (k3 == b1) atomicAdd(&lh[(u3 >> 7) & 0xFFFu], 1u);
    }

    part[threadIdx.x] = wmma_pair_reduce_finish(acc);
    __syncthreads();

    unsigned* g = hist2 + (size_t)slice * 4096;
    for (int i = threadIdx.x; i < 4096; i += NTHREADS) {
        unsigned c = lh[i];
        if (c) atomicAdd(&g[i], c);
    }
    if (threadIdx.x == 0) {
        double s = 0.0;
        for (int i = 0; i < NTHREADS; ++i) s += (double)part[i];
        atomicAdd(&sumsq[slice], s * (1.0 / 16.0));  // undo 16x WMMA redundancy
    }
}

// ---------------- Scan 2: find b2 within bucket b1 -------------------------
__global__ void k_scan2(const unsigned* __restrict__ hist2,
                        const unsigned* __restrict__ need1,
                        unsigned* __restrict__ b2, unsigned* __restrict__ need2) {
    const int s = threadIdx.x;  // 64 threads
    const unsigned* h = hist2 + (size_t)s * 4096;
    const unsigned target = need1[s];
    unsigned cum = 0;
    for (int b = 4095; b >= 0; --b) {
        unsigned c = h[b];
        if (cum + c >= target) { b2[s] = (unsigned)b; need2[s] = target - cum; return; }
        cum += c;
    }
}

// ------- Pass 3: WMMA sum for (b1, mid>b2); low-7-bit hist at (b1,b2) ------
__global__ void __launch_bounds__(NTHREADS)
k_pass3(const float* __restrict__ in, const float* __restrict__ lab,
        const unsigned* __restrict__ b1arr, const unsigned* __restrict__ b2arr,
        unsigned* __restrict__ hist3, double* __restrict__ sumsq) {
    __shared__ unsigned lh[128];
    __shared__ float part[NTHREADS];
    const int slice = blockIdx.x >> 3;
    const int chunk = blockIdx.x & 7;
    const unsigned b1 = b1arr[slice];
    const unsigned b2 = b2arr[slice];
    if (threadIdx.x < 128) lh[threadIdx.x] = 0u;
    __syncthreads();

    const size_t base = (size_t)slice * NELEM + (size_t)chunk * CHUNK;
    const float4* in4 = (const float4*)(in + base);
    const float4* lb4 = (const float4*)(lab + base);

    v8f acc = {0.f, 0.f, 0.f, 0.f, 0.f, 0.f, 0.f, 0.f};
    const v2f ones = {1.0f, 1.0f};

    for (int it = 0; it < ITERS; ++it) {
        const int idx = it * NTHREADS + threadIdx.x;
        float4 a = in4[idx];
        float4 b = lb4[idx];
        float d0 = fabsf(a.x - b.x), d1 = fabsf(a.y - b.y);
        float d2 = fabsf(a.z - b.z), d3 = fabsf(a.w - b.w);
        unsigned u0 = __float_as_uint(d0), u1 = __float_as_uint(d1);
        unsigned u2 = __float_as_uint(d2), u3 = __float_as_uint(d3);
        unsigned k0 = u0 >> 19, k1 = u1 >> 19, k2 = u2 >> 19, k3 = u3 >> 19;
        unsigned m0 = (u0 >> 7) & 0xFFFu, m1 = (u1 >> 7) & 0xFFFu;
        unsigned m2 = (u2 >> 7) & 0xFFFu, m3 = (u3 >> 7) & 0xFFFu;
        v2f p = { (k0 == b1 && m0 > b2) ? d0 * d0 : 0.0f,
                  (k1 == b1 && m1 > b2) ? d1 * d1 : 0.0f };
        v2f q = { (k2 == b1 && m2 > b2) ? d2 * d2 : 0.0f,
                  (k3 == b1 && m3 > b2) ? d3 * d3 : 0.0f };
        acc = __builtin_amdgcn_wmma_f32_16x16x4_f32(false, ones, false, p,
                                                    (short)0, acc, false, false);
        acc = __builtin_amdgcn_wmma_f32_16x16x4_f32(false, ones, false, q,
                                                    (short)0, acc, false, false);
        if (k0 == b1 && m0 == b2) atomicAdd(&lh[u0 & 0x7Fu], 1u);
        if (k1 == b1 && m1 == b2) atomicAdd(&lh[u1 & 0x7Fu], 1u);
        if (k2 == b1 && m2 == b2) atomicAdd(&lh[u2 & 0x7Fu], 1u);
        if (k3 == b1 && m3 == b2) atomicAdd(&lh[u3 & 0x7Fu], 1u);
    }

    part[threadIdx.x] = wmma_pair_reduce_finish(acc);
    __syncthreads();

    unsigned* g = hist3 + (size_t)slice * 128;
    if (threadIdx.x < 128) {
        unsigned c = lh[threadIdx.x];
        if (c) atomicAdd(&g[threadIdx.x], c);
    }
    if (threadIdx.x == 0) {
        double s = 0.0;
        for (int i = 0; i < NTHREADS; ++i) s += (double)part[i];
        atomicAdd(&sumsq[slice], s * (1.0 / 16.0));
    }
}

// ------- Final: exact tie resolution at the cut value + global combine -----
__global__ void k_final(const unsigned* __restrict__ hist3,
                        const unsigned* __restrict__ b1arr,
                        const unsigned* __restrict__ b2arr,
                        const unsigned* __restrict__ need2arr,
                        const double* __restrict__ sumsq,
                        float* __restrict__ out) {
    __shared__ double sh[NSLICE];
    const int s = threadIdx.x;  // 64 threads
    const unsigned* h = hist3 + (size_t)s * 128;
    unsigned rem = need2arr[s];
    const unsigned bb1 = b1arr[s], bb2 = b2arr[s];
    double acc = sumsq[s];
    for (int t = 127; t >= 0 && rem; --t) {
        unsigned c = h[t];
        if (!c) continue;
        unsigned take = (c < rem) ? c : rem;
        unsigned bits = (bb1 << 19) | (bb2 << 7) | (unsigned)t;
        float v = __uint_as_float(bits);
        acc += (double)take * (double)v * (double)v;
        rem -= take;
    }
    sh[s] = acc / (double)KSEL;  // per-slice mean over top-k
    __syncthreads();
    if (s == 0) {
        double r = 0.0;
        for (int i = 0; i < NSLICE; ++i) r += sh[i];
        out[0] = (float)(r / 16.0);  // / B
    }
}

// ---------------------------------------------------------------------------
extern "C" void kernel_launch(void* const* d_in, const int* in_sizes, int n_in,
                              void* d_out, int out_size, void* d_ws, size_t ws_size,
                              hipStream_t stream) {
    (void)in_sizes; (void)n_in; (void)out_size;
    const float* in  = (const float*)d_in[0];
    const float* lab = (const float*)d_in[1];
    float* out = (float*)d_out;

    // Workspace layout (u32 units):
    //   hist1[64*4096] | hist2[64*4096] | hist3[64*128] |
    //   b1[64] need1[64] b2[64] need2[64] | sumsq: double[64]
    unsigned* uws   = (unsigned*)d_ws;
    unsigned* hist1 = uws;
    unsigned* hist2 = hist1 + (size_t)NSLICE * 4096;
    unsigned* hist3 = hist2 + (size_t)NSLICE * 4096;
    unsigned* b1    = hist3 + (size_t)NSLICE * 128;
    unsigned* need1 = b1 + NSLICE;
    unsigned* b2    = need1 + NSLICE;
    unsigned* need2 = b2 + NSLICE;
    double*   sumsq = (double*)(need2 + NSLICE);  // byte offset 2,130,944 (8-aligned)
    const size_t ws_used = (size_t)((char*)(sumsq + NSLICE) - (char*)d_ws);

    hipMemsetAsync(d_ws, 0, ws_used, stream);

    const int nblocks = NSLICE * WG_PER_SLICE;  // 512
    k_hist1<<<nblocks, NTHREADS, 0, stream>>>(in, lab, hist1);
    k_scan1<<<1, NSLICE, 0, stream>>>(hist1, b1, need1);
    k_pass2<<<nblocks, NTHREADS, 0, stream>>>(in, lab, b1, hist2, sumsq);
    k_scan2<<<1, NSLICE, 0, stream>>>(hist2, need1, b2, need2);
    k_pass3<<<nblocks, NTHREADS, 0, stream>>>(in, lab, b1, b2, hist3, sumsq);
    k_final<<<1, NSLICE, 0, stream>>>(hist3, b1, b2, need2, sumsq, out);
}